// EdgeConv_3805341024636
// MI455X (gfx1250) — compile-verified
//
#include <hip/hip_runtime.h>

#define NPTS  8192
#define BATCH 4
#define OUTC  64
#define KNNK  20
#define EPSN  1e-5f
#define SLOPE 0.2f
#define CHUNK 256

typedef __attribute__((ext_vector_type(2))) float v2f;
typedef __attribute__((ext_vector_type(8))) float v8f;

// ---- CDNA5 async global->LDS copy (ASYNCcnt-tracked), per-lane 4 bytes -------------
__device__ __forceinline__ void async_ld_f32(unsigned lds_off, const float* g) {
    unsigned long long ga = (unsigned long long)g;
    asm volatile("global_load_async_to_lds_b32 %0, %1, off"
                 :: "v"(lds_off), "v"(ga) : "memory");
}
__device__ __forceinline__ void wait_async_zero() {
    asm volatile("s_wait_asynccnt 0" ::: "memory");
}
__device__ __forceinline__ unsigned lds_off_of(const void* p) {
    return (unsigned)(size_t)p;      // low 32 bits of a generic LDS pointer = LDS offset
}

// ---------------- zero the moment accumulators (27 used, padded to 32/batch) --------
__global__ void zero_moments_kernel(float* __restrict__ mom) {
    int t = threadIdx.x;
    if (t < BATCH * 32) mom[t] = 0.0f;
}

// ---------------- KNN (async double-buffered LDS tiles) + fused edge moments --------
__global__ __launch_bounds__(256) void knn_moments_kernel(const float* __restrict__ x,
                                                          int* __restrict__ knn,
                                                          float* __restrict__ mom) {
    __shared__ float sbuf[2][3][CHUNK];
    __shared__ float sacc[27];
    const int t = threadIdx.x;
    if (t < 27) sacc[t] = 0.0f;

    int gid = blockIdx.x * 256 + t;                 // covers BATCH*NPTS
    int b = gid / NPTS;                             // uniform per block (256 | NPTS)
    int n = gid % NPTS;
    const float* xb = x + (size_t)b * 3 * NPTS;
    float qx = xb[n], qy = xb[NPTS + n], qz = xb[2 * NPTS + n];

    float bd[KNNK]; int bi[KNNK];
#pragma unroll
    for (int j = 0; j < KNNK; ++j) { bd[j] = 3.4e38f; bi[j] = 0; }

    // prologue: async-stage chunk 0 into buffer 0
    {
        const float* g = xb + t;
        async_ld_f32(lds_off_of(&sbuf[0][0][t]), g);
        async_ld_f32(lds_off_of(&sbuf[0][1][t]), g + NPTS);
        async_ld_f32(lds_off_of(&sbuf[0][2][t]), g + 2 * NPTS);
    }

    for (int c = 0; c < NPTS / CHUNK; ++c) {
        wait_async_zero();          // my chunk-c writes complete
        __syncthreads();            // everyone's chunk-c writes visible; reads of other buf done
        if (c + 1 < NPTS / CHUNK) { // overlap next chunk's copy with this chunk's scan
            int nb = (c + 1) & 1;
            const float* g = xb + (c + 1) * CHUNK + t;
            async_ld_f32(lds_off_of(&sbuf[nb][0][t]), g);
            async_ld_f32(lds_off_of(&sbuf[nb][1][t]), g + NPTS);
            async_ld_f32(lds_off_of(&sbuf[nb][2][t]), g + 2 * NPTS);
        }
        const float* s0 = sbuf[c & 1][0];
        const float* s1 = sbuf[c & 1][1];
        const float* s2 = sbuf[c & 1][2];
        int m0 = c * CHUNK;
        for (int u = 0; u < CHUNK; ++u) {
            float dx = s0[u] - qx, dy = s1[u] - qy, dz = s2[u] - qz;
            float d = dx * dx + dy * dy + dz * dz;
            if (d < bd[KNNK - 1]) {
                bd[KNNK - 1] = d; bi[KNNK - 1] = m0 + u;
#pragma unroll
                for (int j = KNNK - 1; j > 0; --j) {   // one bubble pass, static indexing
                    if (bd[j] < bd[j - 1]) {
                        float td = bd[j]; bd[j] = bd[j - 1]; bd[j - 1] = td;
                        int   ti = bi[j]; bi[j] = bi[j - 1]; bi[j - 1] = ti;
                    }
                }
            }
        }
    }

    // ---- write KNN indices ----
    int* kout = knn + (size_t)gid * KNNK;
#pragma unroll
    for (int j = 0; j < KNNK; ++j) kout[j] = bi[j];

    // ---- fused edge-feature moments (indices still in registers) ----
    float s1v[6], s2v[21];
#pragma unroll
    for (int i = 0; i < 6; ++i)  s1v[i] = 0.0f;
#pragma unroll
    for (int i = 0; i < 21; ++i) s2v[i] = 0.0f;
#pragma unroll
    for (int j = 0; j < KNNK; ++j) {
        int jj = bi[j];
        float e[6];
        e[0] = qx; e[1] = qy; e[2] = qz;
        e[3] = xb[jj] - qx;
        e[4] = xb[NPTS + jj] - qy;
        e[5] = xb[2 * NPTS + jj] - qz;
#pragma unroll
        for (int c2 = 0; c2 < 6; ++c2) s1v[c2] += e[c2];
        int ti = 0;
#pragma unroll
        for (int c2 = 0; c2 < 6; ++c2)
#pragma unroll
            for (int d2 = c2; d2 < 6; ++d2) { s2v[ti] += e[c2] * e[d2]; ++ti; }
    }
#pragma unroll
    for (int c2 = 0; c2 < 6; ++c2) atomicAdd(&sacc[c2], s1v[c2]);
#pragma unroll
    for (int ti = 0; ti < 21; ++ti) atomicAdd(&sacc[6 + ti], s2v[ti]);
    __syncthreads();
    if (t < 27) atomicAdd(&mom[b * 32 + t], sacc[t]);
}

// ---------------- per-(b,o) instance-norm stats through the linear layer ------------
__global__ void stats_kernel(const float* __restrict__ W, const float* __restrict__ mom,
                             float* __restrict__ meanArr, float* __restrict__ invArr) {
    int t = threadIdx.x;            // 256 = BATCH*OUTC
    int b = t >> 6, o = t & 63;
    const float* m = mom + b * 32;
    const float* w = W + o * 6;
    const float invcnt = 1.0f / (float)(NPTS * KNNK);
    float mean = 0.0f;
#pragma unroll
    for (int c = 0; c < 6; ++c) mean += w[c] * m[c];
    mean *= invcnt;
    float e2 = 0.0f;
    int ti = 0;
#pragma unroll
    for (int c = 0; c < 6; ++c)
#pragma unroll
        for (int d = c; d < 6; ++d) {
            float f = (c == d) ? 1.0f : 2.0f;
            e2 += f * w[c] * w[d] * m[6 + ti];
            ++ti;
        }
    e2 *= invcnt;
    float var = e2 - mean * mean;
    meanArr[t] = mean;
    invArr[t]  = rsqrtf(var + EPSN);
}

// ---------------- WMMA conv + norm + LeakyReLU + max-over-k -------------------------
// One wave: 16 points (N-cols) x 64 out-channels (4 M-tiles).
// A = W tile [16x4] f32; B = edge vectors [4x16]; D accumulated over 2 K-chunks.
__global__ __launch_bounds__(256) void edgeconv_wmma_kernel(
        const float* __restrict__ x, const float* __restrict__ W,
        const int* __restrict__ knn, const float* __restrict__ meanArr,
        const float* __restrict__ invArr, float* __restrict__ out) {
    int lane = threadIdx.x & 31;
    int wave = threadIdx.x >> 5;
    int tile = blockIdx.x * 8 + wave;              // 0 .. BATCH*(NPTS/16)-1
    int b    = tile / (NPTS / 16);
    int n0   = (tile % (NPTS / 16)) * 16;
    int col  = lane & 15;
    int hi   = lane >> 4;                          // lanes 16-31 hold K/M upper halves
    int n    = n0 + col;
    const float* xb = x + (size_t)b * 3 * NPTS;

    // A tiles: load all 6 weights unconditionally, then cndmask (no exec divergence)
    v2f a0[4], a1[4];
#pragma unroll
    for (int ot = 0; ot < 4; ++ot) {
        const float* wr = W + (ot * 16 + col) * 6;
        float w0 = wr[0], w1 = wr[1], w2 = wr[2], w3 = wr[3], w4 = wr[4], w5 = wr[5];
        a0[ot].x = hi ? w2 : w0;
        a0[ot].y = hi ? w3 : w1;
        a1[ot].x = hi ? 0.0f : w4;
        a1[ot].y = hi ? 0.0f : w5;
    }
    // C/D layout: lane L, vgpr r -> (M = r + 8*hi, N = col)
    v8f meanv[4], invv[4], runmax[4];
#pragma unroll
    for (int ot = 0; ot < 4; ++ot)
#pragma unroll
        for (int r = 0; r < 8; ++r) {
            int o = ot * 16 + r + 8 * hi;
            meanv[ot][r]  = meanArr[b * 64 + o];
            invv[ot][r]   = invArr[b * 64 + o];
            runmax[ot][r] = -3.4e38f;
        }

    float xn0 = xb[n], xn1 = xb[NPTS + n], xn2 = xb[2 * NPTS + n];
    const int* kn = knn + (size_t)(b * NPTS + n) * KNNK;

    for (int k = 0; k < KNNK; ++k) {
        int j = kn[k];
        float e3 = xb[j] - xn0;
        float e4 = xb[NPTS + j] - xn1;
        float e5 = xb[2 * NPTS + j] - xn2;
        // B: vgpr0 -> K = 0(lo)/2(hi), vgpr1 -> K = 1(lo)/3(hi); chunk1 = ch 4..7 (6,7 zero)
        v2f bm0, bm1;
        bm0.x = hi ? xn2 : xn0;
        bm0.y = hi ? e3  : xn1;
        bm1.x = hi ? 0.0f : e4;
        bm1.y = hi ? 0.0f : e5;
#pragma unroll
        for (int ot = 0; ot < 4; ++ot) {
            v8f c = {};
            c = __builtin_amdgcn_wmma_f32_16x16x4_f32(false, a0[ot], false, bm0,
                                                      (short)0, c, false, false);
            c = __builtin_amdgcn_wmma_f32_16x16x4_f32(false, a1[ot], false, bm1,
                                                      (short)0, c, false, false);
#pragma unroll
            for (int r = 0; r < 8; ++r) {
                float yn = (c[r] - meanv[ot][r]) * invv[ot][r];
                yn = yn >= 0.0f ? yn : SLOPE * yn;
                runmax[ot][r] = fmaxf(runmax[ot][r], yn);
            }
        }
    }
#pragma unroll
    for (int ot = 0; ot < 4; ++ot)
#pragma unroll
        for (int r = 0; r < 8; ++r) {
            int o = ot * 16 + r + 8 * hi;
            out[((size_t)b * OUTC + o) * NPTS + n] = runmax[ot][r];
        }
}

// ------------------------------------------------------------------------------------
extern "C" void kernel_launch(void* const* d_in, const int* in_sizes, int n_in,
                              void* d_out, int out_size, void* d_ws, size_t ws_size,
                              hipStream_t stream) {
    (void)in_sizes; (void)n_in; (void)out_size; (void)ws_size;
    const float* x = (const float*)d_in[0];      // [4,3,8192]
    const float* W = (const float*)d_in[1];      // [64,6]
    float* out = (float*)d_out;                  // [4,64,8192]

    int*   knn     = (int*)d_ws;                                          // B*N*K ints
    size_t knnB    = (size_t)BATCH * NPTS * KNNK * sizeof(int);
    float* mom     = (float*)((char*)d_ws + knnB);                        // B*32 floats
    float* meanArr = mom + BATCH * 32;                                    // B*O
    float* invArr  = meanArr + BATCH * OUTC;                              // B*O

    zero_moments_kernel<<<1, 128, 0, stream>>>(mom);
    knn_moments_kernel<<<BATCH * NPTS / 256, 256, 0, stream>>>(x, knn, mom);
    stats_kernel<<<1, 256, 0, stream>>>(W, mom, meanArr, invArr);
    edgeconv_wmma_kernel<<<BATCH * (NPTS / 16) / 8, 256, 0, stream>>>(
        x, W, knn, meanArr, invArr, out);
}